// HierarchicalTexture_41120016892627
// MI455X (gfx1250) — compile-verified
//
#include <hip/hip_runtime.h>

// Hierarchical texture sampling (4 mip levels, bilinear, border padding,
// align_corners=False), 16 channels, summed over levels.
//
// Layouts (all f32 unless noted):
//   d_in[0] uv    : [1, 2, 512, 512]            gx = uv[pix], gy = uv[HW + pix]
//   d_in[1] texid : [1] int
//   d_in[2] data  : [2, 16, 2048, 1024]
//   d_out         : [1, 16, 512, 512]

#define HW_TOTAL   (512 * 512)
#define N_CH       16
#define ROW_STRIDE 1024                 // floats per texture row
#define CH_STRIDE  (2048 * 1024)        // floats per channel plane (8 MB)
#define TEX_STRIDE (N_CH * CH_STRIDE)   // floats per texture

__global__ __launch_bounds__(256) void hier_tex_sample_kernel(
    const float* __restrict__ uv,
    const int*   __restrict__ texid_p,
    const float* __restrict__ data,
    float*       __restrict__ out)
{
    const int pix = blockIdx.x * blockDim.x + threadIdx.x;

    // Streaming inputs: non-temporal so they don't evict L2-resident texture.
    const float gx = __builtin_nontemporal_load(&uv[pix]);
    const float gy = __builtin_nontemporal_load(&uv[HW_TOTAL + pix]);

    const int tid = *texid_p;                      // uniform -> s_load
    const float* __restrict__ tex = data + (size_t)tid * TEX_STRIDE;

    float acc[N_CH];
#pragma unroll
    for (int c = 0; c < N_CH; ++c) acc[c] = 0.0f;

    int row_off = 0;                               // level row offset: 0,1024,1536,1792
#pragma unroll
    for (int lvl = 0; lvl < 4; ++lvl) {
        const int   w  = 1024 >> lvl;
        const float fw = (float)w;

        // align_corners=False mapping + border clamp (matches reference exactly)
        float ix = ((gx + 1.0f) * fw - 1.0f) * 0.5f;
        float iy = ((gy + 1.0f) * fw - 1.0f) * 0.5f;
        ix = fminf(fmaxf(ix, 0.0f), fw - 1.0f);
        iy = fminf(fmaxf(iy, 0.0f), fw - 1.0f);

        const float fx0 = floorf(ix);
        const float fy0 = floorf(iy);
        const float wx  = ix - fx0;
        const float wy  = iy - fy0;
        const int   x0  = (int)fx0;
        const int   y0  = (int)fy0;
        const int   y1  = (y0 + 1 < w) ? (y0 + 1) : (w - 1);

        // Border handling needs NO selects on the loaded data:
        //   x1 != x0+1 can only happen when ix clamps to exactly w-1, which
        //   forces wx == 0, so the x1 taps get weight 0 (same for y: wy == 0
        //   when y1 would clamp). The extra lane of the b64 load is always a
        //   finite in-allocation value, and 0 * finite == 0.
        const int o0 = (row_off + y0) * ROW_STRIDE + x0;   // row y0, texels x0,x0+1
        const int o1 = (row_off + y1) * ROW_STRIDE + x0;   // row y1, texels x0,x0+1

        const float w00 = (1.0f - wx) * (1.0f - wy);
        const float w01 = wx * (1.0f - wy);
        const float w10 = (1.0f - wx) * wy;
        const float w11 = wx * wy;

        // Uniform per-channel base walk (SGPR adds), divergent 32-bit offsets
        // (GVS addressing with scale_offset). Two 8-byte row-pair gathers and
        // four FMAs per channel; 16 independent accumulation chains for ILP.
        const float* texc = tex;
#pragma unroll
        for (int c = 0; c < N_CH; ++c) {
            float2 ra, rb;
            __builtin_memcpy(&ra, texc + (unsigned)o0, sizeof(float2));
            __builtin_memcpy(&rb, texc + (unsigned)o1, sizeof(float2));
            acc[c] = fmaf(w00, ra.x,
                     fmaf(w01, ra.y,
                     fmaf(w10, rb.x,
                     fmaf(w11, rb.y, acc[c]))));
            texc += CH_STRIDE;
        }

        row_off += w;
    }

    // Streaming output: non-temporal stores, coalesced per channel plane.
#pragma unroll
    for (int c = 0; c < N_CH; ++c)
        __builtin_nontemporal_store(acc[c], &out[(size_t)c * HW_TOTAL + pix]);
}

extern "C" void kernel_launch(void* const* d_in, const int* in_sizes, int n_in,
                              void* d_out, int out_size, void* d_ws, size_t ws_size,
                              hipStream_t stream) {
    (void)in_sizes; (void)n_in; (void)out_size; (void)d_ws; (void)ws_size;

    const float* uv    = (const float*)d_in[0];
    const int*   texid = (const int*)  d_in[1];
    const float* data  = (const float*)d_in[2];
    float*       out   = (float*)d_out;

    const int threads = 256;                   // 8 wave32s per block
    const int blocks  = HW_TOTAL / threads;    // 1024 blocks
    hier_tex_sample_kernel<<<blocks, threads, 0, stream>>>(uv, texid, data, out);
}